// PagedAttention_20916490732004
// MI455X (gfx1250) — compile-verified
//
#include <hip/hip_runtime.h>
#include <math.h>

#define BB 16
#define NUM_HEADS 16
#define HEAD_DIM 64
#define HIDDEN 1024
#define BLOCK_SZ 16
#define MAX_CTX 4096
#define BLOCKS_PER_SEQ 256
#define NCHUNK 8
#define CHUNK (MAX_CTX / NCHUNK)

typedef __attribute__((ext_vector_type(2))) float v2f;
typedef __attribute__((ext_vector_type(8))) float v8f;

// ---------------------------------------------------------------------------
// Y[16 x HIDDEN] = X[16 x HIDDEN] @ W^T  (W is [HIDDEN x HIDDEN], row-major)
// One wave per 16-column tile; K-loop of 256 x V_WMMA_F32_16X16X4_F32.
// A-tile: lane = M (0..15), half-wave picks K pair -> contiguous float2 load.
// B-tile: lane = N, half-wave picks K pair -> contiguous float2 from W row.
// ---------------------------------------------------------------------------
__global__ __launch_bounds__(32) void proj16_wmma(const float* __restrict__ X,
                                                  const float* __restrict__ W,
                                                  float* __restrict__ Y) {
  const int n0   = blockIdx.x * 16;
  const int lane = threadIdx.x & 31;
  const int mh   = lane & 15;              // M for A, N-offset for B
  const int kb   = (lane >> 4) << 1;       // K sub-pair: {0,1} or {2,3}
  const float* xp = X + mh * HIDDEN + kb;
  const float* wp = W + (n0 + mh) * HIDDEN + kb;   // W^T[k][n] == W[n][k]
  v8f c = {};
#pragma unroll 8
  for (int k = 0; k < HIDDEN; k += 4) {
    v2f a = *(const v2f*)(xp + k);
    v2f b = *(const v2f*)(wp + k);
    c = __builtin_amdgcn_wmma_f32_16x16x4_f32(false, a, false, b, (short)0, c,
                                              false, false);
  }
  const int mbase = (lane >> 4) << 3;      // D: m = r + 8*(lane>=16)
  const int n     = n0 + mh;
#pragma unroll
  for (int r = 0; r < 8; ++r)
    Y[(mbase + r) * HIDDEN + n] = c[r];
}

// ---------------------------------------------------------------------------
// Flash-decoding partial pass: one workgroup per (b, h, chunk).
// 4 waves; each wave owns one token at a time: 32 lanes x float2 = the full
// 64-dim head row, fully coalesced. Online softmax per wave, LDS merge.
// New-token K/V come from workspace (inputs must not be mutated).
// Partial record (66 floats): [m, l, acc[64]]
// ---------------------------------------------------------------------------
__global__ __launch_bounds__(128) void attn_partial(
    const float* __restrict__ qws, const float* __restrict__ kws,
    const float* __restrict__ vws, const float* __restrict__ kc,
    const float* __restrict__ vc, const int* __restrict__ btab,
    const int* __restrict__ pos, float* __restrict__ part) {
  const int bh    = blockIdx.x;
  const int b     = bh >> 4;
  const int h     = bh & 15;
  const int chunk = blockIdx.y;
  const int wid   = threadIdx.x >> 5;
  const int lane  = threadIdx.x & 31;
  const int p_b   = pos[b];
  const float scale = 0.125f;              // 1/sqrt(64)

  const int qoff = b * HIDDEN + h * HEAD_DIM + lane * 2;
  const float2 qv = *(const float2*)(qws + qoff);

  float m = -INFINITY, l = 0.f;
  float accx = 0.f, accy = 0.f;

  const int t0 = chunk * CHUNK;
  for (int t = t0 + wid; t < t0 + CHUNK; t += 4) {
    if (t > p_b) break;                    // uniform per wave, tokens ascending
    float2 kv, vv;
    if (t == p_b) {                        // current token: K/V from workspace
      kv = *(const float2*)(kws + qoff);
      vv = *(const float2*)(vws + qoff);
    } else {
      const int  blk  = btab[b * BLOCKS_PER_SEQ + (t >> 4)];
      const long base = ((((long)blk * BLOCK_SZ) + (t & 15)) * NUM_HEADS + h) *
                            HEAD_DIM + lane * 2;
      kv = *(const float2*)(kc + base);
      vv = *(const float2*)(vc + base);
    }
    float s = qv.x * kv.x + qv.y * kv.y;
#pragma unroll
    for (int off = 16; off >= 1; off >>= 1)  // wave32 reduction
      s += __shfl_xor(s, off, 32);
    s *= scale;
    const float mn   = fmaxf(m, s);
    const float corr = __expf(m - mn);       // m=-inf -> 0
    const float pe   = __expf(s - mn);
    l    = l * corr + pe;
    accx = accx * corr + pe * vv.x;
    accy = accy * corr + pe * vv.y;
    m = mn;
  }

  __shared__ float sm[4], sl[4], sacc[4][HEAD_DIM];
  if (lane == 0) { sm[wid] = m; sl[wid] = l; }
  sacc[wid][lane * 2]     = accx;
  sacc[wid][lane * 2 + 1] = accy;
  __syncthreads();

  if (threadIdx.x < HEAD_DIM) {
    const int d = threadIdx.x;
    const float M = fmaxf(fmaxf(sm[0], sm[1]), fmaxf(sm[2], sm[3]));
    float L = 0.f, a = 0.f;
#pragma unroll
    for (int w = 0; w < 4; ++w) {
      const float cw = (sm[w] == -INFINITY) ? 0.f : __expf(sm[w] - M);
      L += sl[w] * cw;
      a += sacc[w][d] * cw;
    }
    float* pp = part + (bh * NCHUNK + chunk) * 66;
    if (d == 0) { pp[0] = M; pp[1] = L; }
    pp[2 + d] = a;
  }
}

// ---------------------------------------------------------------------------
// LSE-combine the NCHUNK partials per (b,h) and normalize.
// ---------------------------------------------------------------------------
__global__ __launch_bounds__(64) void attn_combine(const float* __restrict__ part,
                                                   float* __restrict__ aout) {
  const int bh = blockIdx.x;
  const int b  = bh >> 4, h = bh & 15;
  const int d  = threadIdx.x;
  float M = -INFINITY;
#pragma unroll
  for (int c = 0; c < NCHUNK; ++c)
    M = fmaxf(M, part[(bh * NCHUNK + c) * 66]);
  float L = 0.f, a = 0.f;
#pragma unroll
  for (int c = 0; c < NCHUNK; ++c) {
    const float* pp = part + (bh * NCHUNK + c) * 66;
    const float mm = pp[0];
    const float cw = (mm == -INFINITY) ? 0.f : __expf(mm - M);
    L += pp[1] * cw;
    a += pp[2 + d] * cw;
  }
  aout[b * HIDDEN + h * HEAD_DIM + d] = a / L;  // token 0 always valid -> L>0
}

// ---------------------------------------------------------------------------
extern "C" void kernel_launch(void* const* d_in, const int* in_sizes, int n_in,
                              void* d_out, int out_size, void* d_ws,
                              size_t ws_size, hipStream_t stream) {
  const float* hidden = (const float*)d_in[0];
  const float* kc     = (const float*)d_in[1];
  const float* vc     = (const float*)d_in[2];
  const int*   btab   = (const int*)d_in[3];
  const int*   pos    = (const int*)d_in[4];
  const float* Wq     = (const float*)d_in[5];
  const float* Wk     = (const float*)d_in[6];
  const float* Wv     = (const float*)d_in[7];
  const float* Wo     = (const float*)d_in[8];

  float* ws   = (float*)d_ws;
  float* qws  = ws;                               // 16*1024
  float* kws  = ws + 16384;                       // 16*1024
  float* vws  = ws + 32768;                       // 16*1024
  float* part = ws + 49152;                       // 256*8*66
  float* aout = ws + 49152 + BB * NUM_HEADS * NCHUNK * 66;

  proj16_wmma<<<dim3(HIDDEN / 16), 32, 0, stream>>>(hidden, Wq, qws);
  proj16_wmma<<<dim3(HIDDEN / 16), 32, 0, stream>>>(hidden, Wk, kws);
  proj16_wmma<<<dim3(HIDDEN / 16), 32, 0, stream>>>(hidden, Wv, vws);
  attn_partial<<<dim3(BB * NUM_HEADS, NCHUNK), 128, 0, stream>>>(
      qws, kws, vws, kc, vc, btab, pos, part);
  attn_combine<<<dim3(BB * NUM_HEADS), 64, 0, stream>>>(part, aout);
  proj16_wmma<<<dim3(HIDDEN / 16), 32, 0, stream>>>(aout, Wo, (float*)d_out);
}